// SymptomNet_42700564857079
// MI455X (gfx1250) — compile-verified
//
#include <hip/hip_runtime.h>
#include <stdint.h>

typedef __attribute__((ext_vector_type(2))) float v2f;
typedef __attribute__((ext_vector_type(4))) float v4f;
typedef __attribute__((ext_vector_type(8))) float v8f;

constexpr int kB = 64;      // batches
constexpr int kN = 8192;    // patches per batch
constexpr int kD = 128;     // feature dim
constexpr int kM = 64;      // prototypes
constexpr int kNT = 128;    // n-rows per workgroup (8 waves x 16)
constexpr float kAlpha = 0.1f;
constexpr float kEps = 1e-10f;

__device__ __forceinline__ unsigned long long umin64(unsigned long long a,
                                                     unsigned long long b) {
  return a < b ? a : b;
}

__device__ __forceinline__ unsigned long long shfl_xor_u64(unsigned long long v,
                                                           int mask) {
  unsigned lo = (unsigned)v;
  unsigned hi = (unsigned)(v >> 32);
  lo = __shfl_xor((int)lo, mask, 32);
  hi = __shfl_xor((int)hi, mask, 32);
  return ((unsigned long long)hi << 32) | lo;
}

// Re-arm per-(b,m) min keys each launch (harness does not re-poison d_ws).
__global__ void init_ws_kernel(unsigned long long* __restrict__ ws) {
  int i = blockIdx.x * blockDim.x + threadIdx.x;
  if (i < kB * kM) ws[i] = ~0ull;
}

// One workgroup = 8 waves, covers 128 n-rows x all 64 prototypes for one batch.
// Each wave owns a 16(n) x 64(m) strip: 4 WMMA accumulators (16x16 f32 each).
// K is processed 8 at a time with one b128 load per operand: the lane->k
// mapping of a WMMA quad is permutable as long as A and B agree, so lane-half
// h supplies k = k0+4h..k0+4h+3 and we issue two V_WMMA_F32_16X16X4_F32
// (.xy sub-fragment, then .zw) per 8 k-values.
__global__ __launch_bounds__(256, 1)
void proto_min_kernel(const float* __restrict__ x, const float* __restrict__ pv,
                      unsigned long long* __restrict__ ws) {
  __shared__ __align__(16) float lds_pv[kM * kD];  // 32 KB prototypes (reused 8192x)
  __shared__ float lds_p2[kM];                     // ||p||^2
  __shared__ float lds_x2[kNT];                    // ||x||^2 per block row
  __shared__ unsigned long long lds_min[kM];       // cross-wave min keys

  const int tid  = threadIdx.x;
  const int wave = tid >> 5;
  const int lane = tid & 31;
  const int half = lane >> 4;   // which 16-lane half of the wave32
  const int col  = lane & 15;
  const int bat  = blockIdx.y;
  const int n_base = blockIdx.x * kNT;

  // Cooperative pv -> LDS (b128, fully coalesced; 2048 loads per workgroup).
  {
    const v4f* pv4 = (const v4f*)pv;
    v4f* lpv4 = (v4f*)lds_pv;
    #pragma unroll
    for (int i = tid; i < kM * kD / 4; i += 256) lpv4[i] = pv4[i];
  }
  if (tid < kM) lds_min[tid] = ~0ull;
  __syncthreads();

  // ||p||^2 per prototype (one thread per prototype, reads LDS).
  if (tid < kM) {
    float s = 0.f;
    #pragma unroll 8
    for (int d = 0; d < kD; ++d) {
      float p = lds_pv[tid * kD + d];
      s = fmaf(p, p, s);
    }
    lds_p2[tid] = s;
  }

  // ---- GEMM: out[n,m] = sum_k x[n,k] * pv[m,k] ----
  const int row_loc = wave * 16 + col;            // row within block tile
  const float* xrow = x + ((size_t)bat * kN + (n_base + row_loc)) * kD;
  const float* pvb = &lds_pv[col * kD];           // +0/8KB/16KB/24KB for m-tiles
  const int koff = 4 * half;                      // permuted k assignment

  v8f acc0 = {}, acc1 = {}, acc2 = {}, acc3 = {};
  float x2 = 0.f;
  #pragma unroll 2
  for (int k = 0; k < kD; k += 8) {
    v4f a  = *(const v4f*)(xrow + k + koff);              // global_load_b128
    x2 = fmaf(a.x, a.x, fmaf(a.y, a.y, x2));
    x2 = fmaf(a.z, a.z, fmaf(a.w, a.w, x2));
    v4f b0 = *(const v4f*)(pvb + 0 * 16 * kD + k + koff); // ds_load_b128 x4
    v4f b1 = *(const v4f*)(pvb + 1 * 16 * kD + k + koff);
    v4f b2 = *(const v4f*)(pvb + 2 * 16 * kD + k + koff);
    v4f b3 = *(const v4f*)(pvb + 3 * 16 * kD + k + koff);

    v2f alo = __builtin_shufflevector(a, a, 0, 1);
    v2f ahi = __builtin_shufflevector(a, a, 2, 3);
    acc0 = __builtin_amdgcn_wmma_f32_16x16x4_f32(false, alo, false, __builtin_shufflevector(b0, b0, 0, 1), (short)0, acc0, false, false);
    acc1 = __builtin_amdgcn_wmma_f32_16x16x4_f32(false, alo, false, __builtin_shufflevector(b1, b1, 0, 1), (short)0, acc1, false, false);
    acc2 = __builtin_amdgcn_wmma_f32_16x16x4_f32(false, alo, false, __builtin_shufflevector(b2, b2, 0, 1), (short)0, acc2, false, false);
    acc3 = __builtin_amdgcn_wmma_f32_16x16x4_f32(false, alo, false, __builtin_shufflevector(b3, b3, 0, 1), (short)0, acc3, false, false);
    acc0 = __builtin_amdgcn_wmma_f32_16x16x4_f32(false, ahi, false, __builtin_shufflevector(b0, b0, 2, 3), (short)0, acc0, false, false);
    acc1 = __builtin_amdgcn_wmma_f32_16x16x4_f32(false, ahi, false, __builtin_shufflevector(b1, b1, 2, 3), (short)0, acc1, false, false);
    acc2 = __builtin_amdgcn_wmma_f32_16x16x4_f32(false, ahi, false, __builtin_shufflevector(b2, b2, 2, 3), (short)0, acc2, false, false);
    acc3 = __builtin_amdgcn_wmma_f32_16x16x4_f32(false, ahi, false, __builtin_shufflevector(b3, b3, 2, 3), (short)0, acc3, false, false);
  }

  // Lane L covers d mod 8 in {0..3}, lane L^16 covers {4..7} of the same row.
  x2 += __shfl_xor(x2, 16, 32);
  if (half == 0) lds_x2[row_loc] = x2;
  __syncthreads();  // lds_x2 + lds_p2 now valid everywhere

  // ---- Epilogue: dist^2 = x2 + p2 - 2*xp; pack & min-reduce over n ----
  // C/D layout: VGPR v, lane L -> n = v + 8*(L>>4), m-col = L&15.
  const v8f accs[4] = {acc0, acc1, acc2, acc3};
  #pragma unroll
  for (int mt = 0; mt < 4; ++mt) {
    const int m = mt * 16 + col;
    const float p2 = lds_p2[m];
    unsigned long long best = ~0ull;
    #pragma unroll
    for (int v = 0; v < 8; ++v) {
      const int nloc = wave * 16 + v + 8 * half;
      const float d2 = fmaxf(lds_x2[nloc] + p2 - 2.0f * accs[mt][v], 0.0f);
      // non-negative f32 bits are monotonic; low 32 bits = n (first-min tiebreak)
      const unsigned long long key =
          ((unsigned long long)__float_as_uint(d2) << 32) |
          (unsigned)(n_base + nloc);
      best = umin64(best, key);
    }
    best = umin64(best, shfl_xor_u64(best, 16));  // fold the two n-halves
    if (half == 0) atomicMin(&lds_min[m], best);  // ds_min_u64 across 8 waves
  }
  __syncthreads();

  if (tid < kM) atomicMin(&ws[(size_t)bat * kM + tid], lds_min[tid]);  // global_atomic_min_u64
}

// One block per batch, one thread per prototype: sqrt/exp + per-batch argmin.
__global__ void finalize_kernel(const unsigned long long* __restrict__ ws,
                                float* __restrict__ out) {
  __shared__ unsigned long long red[kM];
  const int b = blockIdx.x;
  const int m = threadIdx.x;

  const unsigned long long key = ws[(size_t)b * kM + m];
  const float d2 = __uint_as_float((unsigned)(key >> 32));
  const float dist = sqrtf(d2);
  out[(size_t)b * kM + m] = __expf(kEps - kAlpha * dist);

  // argmin over m (ties -> smallest m, matching jnp.argmin)
  red[m] = ((key >> 32) << 32) | (unsigned)m;
  __syncthreads();
  #pragma unroll
  for (int s = 32; s > 0; s >>= 1) {
    if (m < s) red[m] = umin64(red[m], red[m + s]);
    __syncthreads();
  }
  if (m == 0) {
    const int mwin = (int)(unsigned)(red[0] & 0xFFFFFFFFu);
    const unsigned npos = (unsigned)(ws[(size_t)b * kM + mwin] & 0xFFFFFFFFu);
    out[kB * kM + b] = (float)npos;   // min_pos, after max_sim block
  }
}

extern "C" void kernel_launch(void* const* d_in, const int* in_sizes, int n_in,
                              void* d_out, int out_size, void* d_ws, size_t ws_size,
                              hipStream_t stream) {
  const float* x  = (const float*)d_in[0];   // [64, 8192, 128] f32
  const float* pv = (const float*)d_in[1];   // [64, 128] f32
  float* out = (float*)d_out;                // 4096 max_sim + 64 min_pos
  unsigned long long* ws = (unsigned long long*)d_ws;  // 64*64 u64 min keys

  init_ws_kernel<<<dim3((kB * kM + 255) / 256), 256, 0, stream>>>(ws);

  dim3 grid(kN / kNT, kB);   // 64 n-chunks x 64 batches = 4096 workgroups
  proto_min_kernel<<<grid, 256, 0, stream>>>(x, pv, ws);

  finalize_kernel<<<kB, kM, 0, stream>>>(ws, out);
}